// GINEConv_81157702025494
// MI455X (gfx1250) — compile-verified
//
#include <hip/hip_runtime.h>
#include <stdint.h>

typedef float v4f __attribute__((ext_vector_type(4)));
typedef unsigned int u32;
typedef long long s64;

#define D   128
#define DV  (D/4)     // float4 per feature row
#define CAP 64        // adjacency ids staged in LDS per node (avg degree = 16)
#define WPB 8         // waves (nodes) per block in aggregate

__device__ __forceinline__ v4f relu4(v4f t) {
  t.x = fmaxf(t.x, 0.f); t.y = fmaxf(t.y, 0.f);
  t.z = fmaxf(t.z, 0.f); t.w = fmaxf(t.w, 0.f);
  return t;
}

// ---------------- CSR build ----------------

extern "C" __global__ void zero_u32(u32* __restrict__ p, int n) {
  int i = blockIdx.x * blockDim.x + threadIdx.x;
  if (i < n) p[i] = 0u;
}

extern "C" __global__ void hist_kernel(const s64* __restrict__ ei, u32* __restrict__ cnt, int E) {
  int e = blockIdx.x * blockDim.x + threadIdx.x;
  if (e < E) {
    int d = (int)ei[(size_t)E + e];   // dst row of edge_index
    atomicAdd(&cnt[d], 1u);
  }
}

// Single-block exclusive scan over n counts (n = 50000 -> trivial runtime).
// Writes offsets[0..n] and resets cursor[v] = offsets[v] for the fill pass.
extern "C" __global__ void scan_kernel(u32* __restrict__ cursor, u32* __restrict__ offsets, int n) {
  __shared__ u32 part[1024];
  const int t = threadIdx.x;
  const int chunk = (n + 1023) >> 10;
  int base = t * chunk;
  int end  = base + chunk; if (end > n) end = n;
  u32 s = 0;
  for (int i = base; i < end; i++) s += cursor[i];
  part[t] = s;
  __syncthreads();
  for (int off = 1; off < 1024; off <<= 1) {
    u32 vv = (t >= off) ? part[t - off] : 0u;
    __syncthreads();
    part[t] += vv;
    __syncthreads();
  }
  u32 run = part[t] - s;                      // exclusive prefix of this chunk
  for (int i = base; i < end; i++) {
    u32 c = cursor[i];
    offsets[i] = run;
    cursor[i]  = run;                         // start position for fill phase
    run += c;
  }
  if (t == 1023) offsets[n] = part[1023];
}

extern "C" __global__ void fill_kernel(const s64* __restrict__ ei, u32* __restrict__ cursor,
                                       u32* __restrict__ eids, int E) {
  int e = blockIdx.x * blockDim.x + threadIdx.x;
  if (e < E) {
    int d = (int)ei[(size_t)E + e];
    u32 pos = atomicAdd(&cursor[d], 1u);
    eids[pos] = (u32)e;
  }
}

// ---------------- aggregate: one wave32 per node ----------------

extern "C" __global__ __launch_bounds__(WPB * 32) void aggregate_kernel(
    const v4f* __restrict__ nf, const s64* __restrict__ ei, const v4f* __restrict__ ef,
    const float* __restrict__ eps, const u32* __restrict__ offsets,
    const u32* __restrict__ eids, v4f* __restrict__ out, int n)
{
  __shared__ u32 idbuf[WPB * CAP];
  const int wave = threadIdx.x >> 5;
  const int lane = threadIdx.x & 31;
  const int v = blockIdx.x * WPB + wave;
  if (v >= n) return;                          // uniform per wave: EXEC stays all-1s

  const u32 beg = offsets[v];
  const u32 end = offsets[v + 1];
  const u32 deg = end - beg;
  const int wbase = wave * CAP;

  // CDNA5 async global->LDS: stage up to CAP adjacency ids into a wave-private
  // LDS slice (eids has CAP ints of tail slack, so unconditional loads are safe).
  u32 lds0 = (u32)(uintptr_t)(&idbuf[wbase + lane]);
  unsigned long long ga = (unsigned long long)(uintptr_t)(eids + beg + lane);
  asm volatile("global_load_async_to_lds_b32 %0, %1, off"            :: "v"(lds0), "v"(ga) : "memory");
  asm volatile("global_load_async_to_lds_b32 %0, %1, off offset:128" :: "v"(lds0), "v"(ga) : "memory");

  // Overlap with the async transfer: (1+eps) * node_feat[v]
  const float scale = 1.0f + eps[0];
  v4f acc = nf[(size_t)v * DV + lane] * scale;

  asm volatile("s_wait_asynccnt 0" ::: "memory");

  const u32 m = deg < CAP ? deg : CAP;

  // 1-deep software pipeline over the uniform (e, src) chain, kept on the
  // scalar pipe via readfirstlane so the vector pipe only does row traffic.
  u32 e_cur = 0; int s_cur = 0;
  if (m > 0) {
    e_cur = __builtin_amdgcn_readfirstlane(idbuf[wbase]);
    s_cur = (int)ei[e_cur];                    // s_load_b64 (uniform)
  }
  for (u32 k = 0; k < m; k++) {
    const u32 e = e_cur;
    const int s = s_cur;
    if (k + 1 < m) {                           // prefetch next uniform indices
      e_cur = __builtin_amdgcn_readfirstlane(idbuf[wbase + k + 1]);
      s_cur = (int)ei[e_cur];
    }
    v4f a = nf[(size_t)s * DV + lane];         // L2-resident gather (keep temporal)
    v4f b = __builtin_nontemporal_load(ef + (size_t)e * DV + lane);  // streamed once
    acc += relu4(a + b);
  }
  for (u32 k = CAP; k < deg; k++) {            // rare tail (deg > 64)
    u32 e = __builtin_amdgcn_readfirstlane(eids[beg + k]);
    int s = (int)ei[e];
    v4f a = nf[(size_t)s * DV + lane];
    v4f b = __builtin_nontemporal_load(ef + (size_t)e * DV + lane);
    acc += relu4(a + b);
  }
  out[(size_t)v * DV + lane] = acc;
}

// ---------------- fallback (ws too small): hardware f32 atomics ----------------

extern "C" __global__ void init_out_kernel(const v4f* __restrict__ nf, const float* __restrict__ eps,
                                           v4f* __restrict__ out, int total) {
  int i = blockIdx.x * blockDim.x + threadIdx.x;
  if (i < total) out[i] = nf[i] * (1.0f + eps[0]);
}

extern "C" __global__ void scatter_kernel(const v4f* __restrict__ nf, const s64* __restrict__ ei,
                                          const v4f* __restrict__ ef, float* __restrict__ out, int E) {
  int gid = blockIdx.x * blockDim.x + threadIdx.x;
  int e = gid >> 5, lane = gid & 31;
  if (e >= E) return;
  int s = (int)ei[e];
  int d = (int)ei[(size_t)E + e];
  v4f a = nf[(size_t)s * DV + lane];
  v4f b = __builtin_nontemporal_load(ef + (size_t)e * DV + lane);
  v4f t = relu4(a + b);
  float* dst = out + (size_t)d * D + lane * 4;
  __hip_atomic_fetch_add(dst + 0, t.x, __ATOMIC_RELAXED, __HIP_MEMORY_SCOPE_AGENT);
  __hip_atomic_fetch_add(dst + 1, t.y, __ATOMIC_RELAXED, __HIP_MEMORY_SCOPE_AGENT);
  __hip_atomic_fetch_add(dst + 2, t.z, __ATOMIC_RELAXED, __HIP_MEMORY_SCOPE_AGENT);
  __hip_atomic_fetch_add(dst + 3, t.w, __ATOMIC_RELAXED, __HIP_MEMORY_SCOPE_AGENT);
}

// ---------------- launch ----------------

extern "C" void kernel_launch(void* const* d_in, const int* in_sizes, int n_in,
                              void* d_out, int out_size, void* d_ws, size_t ws_size,
                              hipStream_t stream)
{
  const float* node = (const float*)d_in[0];
  const s64*   ei   = (const s64*)d_in[1];   // edge_index is int64 (2, E)
  const float* ef   = (const float*)d_in[2];
  const float* eps  = (const float*)d_in[3];
  float* out = (float*)d_out;

  const int n = in_sizes[0] / D;
  const int E = in_sizes[2] / D;

  auto align256 = [](size_t x) { return (x + 255) & ~(size_t)255; };
  const size_t o_cursor = 0;
  const size_t o_off    = align256((size_t)n * 4);
  const size_t o_eids   = o_off + align256(((size_t)n + 1) * 4);
  const size_t need     = o_eids + align256(((size_t)E + CAP) * 4);

  if (ws_size >= need) {
    u32* cursor  = (u32*)((char*)d_ws + o_cursor);
    u32* offsets = (u32*)((char*)d_ws + o_off);
    u32* eids    = (u32*)((char*)d_ws + o_eids);

    zero_u32 <<<(n + 255) / 256, 256, 0, stream>>>(cursor, n);
    hist_kernel<<<(E + 255) / 256, 256, 0, stream>>>(ei, cursor, E);
    scan_kernel<<<1, 1024, 0, stream>>>(cursor, offsets, n);
    fill_kernel<<<(E + 255) / 256, 256, 0, stream>>>(ei, cursor, eids, E);
    aggregate_kernel<<<(n + WPB - 1) / WPB, WPB * 32, 0, stream>>>(
        (const v4f*)node, ei, (const v4f*)ef, eps, offsets, eids, (v4f*)out, n);
  } else {
    const int total = n * DV;
    init_out_kernel<<<(total + 255) / 256, 256, 0, stream>>>((const v4f*)node, eps, (v4f*)out, total);
    const long long tt = (long long)E * 32;
    scatter_kernel<<<(int)((tt + 255) / 256), 256, 0, stream>>>((const v4f*)node, ei, (const v4f*)ef, out, E);
  }
}